// QMIExtendedParticleQuantumEncoder_33380485825084
// MI455X (gfx1250) — compile-verified
//
#include <hip/hip_runtime.h>
#include <math.h>

// ---------------------------------------------------------------------------
// Quantum encoder: blocks 1-3 per-thread (register statevector), block 4
// (10 qubits) one wave32 per sample, state in LDS, single-qubit gate sweeps
// fused into 16x16 subspace unitaries applied with V_WMMA_F32_16X16X4_F32.
// ---------------------------------------------------------------------------

typedef float v2f __attribute__((ext_vector_type(2)));
typedef float v8f __attribute__((ext_vector_type(8)));

struct C2 { float r, i; };
struct M2 { C2 m[2][2]; };

__device__ __forceinline__ C2 cmul(C2 a, C2 b) {
    C2 o; o.r = a.r * b.r - a.i * b.i; o.i = a.r * b.i + a.i * b.r; return o;
}

// state <- RZ(phi) * RY(pt) * RX(eta) * state   (reference applies RX,RY,RZ)
__device__ __forceinline__ M2 enc_gate(float eta, float pt, float phi) {
    float cx = __cosf(eta * 0.5f), sx = __sinf(eta * 0.5f);
    float cy = __cosf(pt  * 0.5f), sy = __sinf(pt  * 0.5f);
    float cz = __cosf(phi * 0.5f), sz = __sinf(phi * 0.5f);
    // T = RY * RX
    C2 t00 = { cy * cx,  sy * sx };
    C2 t01 = { -sy * cx, -cy * sx };
    C2 t10 = { sy * cx,  -cy * sx };
    C2 t11 = { cy * cx,  -sy * sx };
    C2 e0 = { cz, -sz }, e1 = { cz, sz };   // RZ row phases
    M2 g;
    g.m[0][0] = cmul(e0, t00); g.m[0][1] = cmul(e0, t01);
    g.m[1][0] = cmul(e1, t10); g.m[1][1] = cmul(e1, t11);
    return g;
}

// state <- RY(ay) * RX(ax) * state
__device__ __forceinline__ M2 layer_gate(float ax, float ay) {
    float cx = __cosf(ax * 0.5f), sx = __sinf(ax * 0.5f);
    float cy = __cosf(ay * 0.5f), sy = __sinf(ay * 0.5f);
    M2 g;
    g.m[0][0] = { cy * cx,  sy * sx };
    g.m[0][1] = { -sy * cx, -cy * sx };
    g.m[1][0] = { sy * cx,  -cy * sx };
    g.m[1][1] = { cy * cx,  -sy * sx };
    return g;
}

__device__ __forceinline__ M2 ident_gate() {
    M2 g;
    g.m[0][0] = { 1.f, 0.f }; g.m[0][1] = { 0.f, 0.f };
    g.m[1][0] = { 0.f, 0.f }; g.m[1][1] = { 1.f, 0.f };
    return g;
}

// ------------------------- block 4: WMMA machinery -------------------------

__device__ __forceinline__ int scat4(int k, const int gb[4]) {
    return (((k >> 3) & 1) << gb[0]) | (((k >> 2) & 1) << gb[1]) |
           (((k >> 1) & 1) << gb[2]) | ((k & 1) << gb[3]);
}
__device__ __forceinline__ int scat6(int m, const int rb[6]) {
    return (((m >> 5) & 1) << rb[0]) | (((m >> 4) & 1) << rb[1]) |
           (((m >> 3) & 1) << rb[2]) | (((m >> 2) & 1) << rb[3]) |
           (((m >> 1) & 1) << rb[4]) | ((m & 1) << rb[5]);
}

// U[n,k] = prod_w g[w].m[n_w][k_w]  (g[0] acts on the highest group bit)
__device__ __forceinline__ C2 umat_entry(const M2 g[4], int n, int k) {
    C2 v = g[0].m[(n >> 3) & 1][(k >> 3) & 1];
    v = cmul(v, g[1].m[(n >> 2) & 1][(k >> 2) & 1]);
    v = cmul(v, g[2].m[(n >> 1) & 1][(k >> 1) & 1]);
    v = cmul(v, g[3].m[n & 1][k & 1]);
    return v;
}

__device__ __forceinline__ v2f make2(float a, float b) {
    v2f v; v[0] = a; v[1] = b; return v;
}

// Apply a 16x16 complex unitary on the 4 state-index bits gb[] (rest bits rb[]).
// S'(64x16) = S(64x16) * U^T, done as 4 row-tiles of complex M16 N16 K16,
// each = 16 chained V_WMMA_F32_16X16X4_F32.
__device__ __forceinline__ void apply_group(float* sR, float* sI, const M2 g[4],
                                            const int gb[4], const int rb[6],
                                            int lane) {
    const int n  = lane & 15;
    const int kh = (lane >> 4) & 1;

    // Build B = U^T tiles: B[k,n] = U[n,k].  Lane holds rows (4t+2kh, 4t+2kh+1), col n.
    v2f br[4], bi[4], nbi[4];
#pragma unroll
    for (int t = 0; t < 4; ++t) {
        int k0 = 4 * t + 2 * kh;
        C2 u0 = umat_entry(g, n, k0);
        C2 u1 = umat_entry(g, n, k0 + 1);
        br[t]  = make2(u0.r, u1.r);
        bi[t]  = make2(u0.i, u1.i);
        nbi[t] = make2(-u0.i, -u1.i);
    }
    const int dk = 1 << gb[3];

#pragma unroll
    for (int rt = 0; rt < 4; ++rt) {
        // A tiles: lane holds S[m, k0], S[m, k0+1] with m = 16*rt + (lane&15)
        v2f ar[4], ai[4];
        int baseA = scat6(16 * rt + n, rb);
#pragma unroll
        for (int t = 0; t < 4; ++t) {
            int addr = baseA + scat4(4 * t + 2 * kh, gb);
            ar[t] = make2(sR[addr], sR[addr + dk]);
            ai[t] = make2(sI[addr], sI[addr + dk]);
        }
        v8f dr = { 0.f, 0.f, 0.f, 0.f, 0.f, 0.f, 0.f, 0.f };
        v8f di = { 0.f, 0.f, 0.f, 0.f, 0.f, 0.f, 0.f, 0.f };
#pragma unroll
        for (int t = 0; t < 4; ++t)
            dr = __builtin_amdgcn_wmma_f32_16x16x4_f32(false, ar[t], false, br[t],
                                                       (short)0, dr, false, false);
#pragma unroll
        for (int t = 0; t < 4; ++t)
            dr = __builtin_amdgcn_wmma_f32_16x16x4_f32(false, ai[t], false, nbi[t],
                                                       (short)0, dr, false, false);
#pragma unroll
        for (int t = 0; t < 4; ++t)
            di = __builtin_amdgcn_wmma_f32_16x16x4_f32(false, ar[t], false, bi[t],
                                                       (short)0, di, false, false);
#pragma unroll
        for (int t = 0; t < 4; ++t)
            di = __builtin_amdgcn_wmma_f32_16x16x4_f32(false, ai[t], false, br[t],
                                                       (short)0, di, false, false);

        // D layout: VGPR j -> row 16*rt + j + 8*kh, col n
        int coln = scat4(n, gb);
#pragma unroll
        for (int j = 0; j < 8; ++j) {
            int addr = scat6(16 * rt + j + 8 * kh, rb) + coln;
            sR[addr] = dr[j];
            sI[addr] = di[j];
        }
    }
}

// CNOT chain wires k->k+1, k = 0..8  (wire w <-> bit 9-w), done as one
// composite basis permutation: final[j] = s0[sig0(sig1(...sig8(j)))]
__device__ __forceinline__ void cnot_chain10(float* sR, float* sI, int lane) {
    float tr[32], ti[32];
    const int base = lane * 32;
#pragma unroll
    for (int t = 0; t < 32; ++t) {
        int src = base + t;
#pragma unroll
        for (int k = 8; k >= 0; --k) {
            int cb = 9 - k, tb = 8 - k;
            src ^= ((src >> cb) & 1) << tb;
        }
        tr[t] = sR[src];
        ti[t] = sI[src];
    }
#pragma unroll
    for (int t = 0; t < 32; ++t) { sR[base + t] = tr[t]; sI[base + t] = ti[t]; }
}

__device__ __forceinline__ void measure10(const float* sR, const float* sI,
                                          int lane, float* out_row) {
    float z[10], xx[10];
#pragma unroll
    for (int w = 0; w < 10; ++w) { z[w] = 0.f; xx[w] = 0.f; }
    const int base = lane * 32;
    float lr[32], li[32];
#pragma unroll
    for (int t = 0; t < 32; ++t) { lr[t] = sR[base + t]; li[t] = sI[base + t]; }
#pragma unroll
    for (int t = 0; t < 32; ++t) {
        int j = base + t;
        float p = lr[t] * lr[t] + li[t] * li[t];
#pragma unroll
        for (int w = 0; w < 10; ++w)
            z[w] += ((j >> (9 - w)) & 1) ? -p : p;
    }
#pragma unroll
    for (int w = 0; w < 10; ++w) {
        int mask = 1 << (9 - w);
#pragma unroll
        for (int t = 0; t < 32; ++t) {
            int j = base + t;
            if (!(j & mask)) {
                int pj = j | mask;   // Re(conj(a_j) * a_pj)
                xx[w] += lr[t] * sR[pj] + li[t] * sI[pj];
            }
        }
    }
#pragma unroll
    for (int w = 0; w < 10; ++w) {
        float zv = z[w], xv = xx[w];
        for (int off = 16; off > 0; off >>= 1) {
            zv += __shfl_xor(zv, off, 32);
            xv += __shfl_xor(xv, off, 32);
        }
        z[w] = zv; xx[w] = xv;
    }
    if (lane == 0) {
#pragma unroll
        for (int w = 0; w < 10; ++w) {
            out_row[12 + w] = z[w];
            out_row[22 + w] = 2.f * xx[w];
        }
    }
}

__global__ void __launch_bounds__(128)
qenc_block4(const float* __restrict__ x, const float* __restrict__ w4,
            float* __restrict__ out) {
    __shared__ float shR[4][1024];
    __shared__ float shI[4][1024];
    const int wave = threadIdx.x >> 5;
    const int lane = threadIdx.x & 31;
    const int b = blockIdx.x * 4 + wave;
    float* sR = shR[wave];
    float* sI = shI[wave];

    const int base = lane * 32;
#pragma unroll
    for (int t = 0; t < 32; ++t) { sR[base + t] = 0.f; sI[base + t] = 0.f; }
    if (lane == 0) sR[0] = 1.f;     // |0...0>

    const int PT4[10]  = { 3, 33, 31,  2, 0, 14, 30, 26, 29, 27 };
    const int ETA4[10] = { 7, 43, 41,  6, 0, 18, 40, 36, 39, 37 };
    const int PHI4[10] = { 11, 53, 51, 10, 1, 22, 50, 46, 49, 47 };
    const float* xb = x + b * 56;

    // wire w <-> state bit (9-w). Groups: wires 0-3, 4-7, 8-9 (identity-padded)
    const int gbA[4] = { 9, 8, 7, 6 }; const int rbA[6] = { 5, 4, 3, 2, 1, 0 };
    const int gbB[4] = { 5, 4, 3, 2 }; const int rbB[6] = { 9, 8, 7, 6, 1, 0 };
    const int gbC[4] = { 3, 2, 1, 0 }; const int rbC[6] = { 9, 8, 7, 6, 5, 4 };

    M2 g[4];
    // ---- encode: per-wire RZ*RY*RX, grouped ----
#pragma unroll
    for (int w = 0; w < 4; ++w)
        g[w] = enc_gate(xb[ETA4[w]], xb[PT4[w]], xb[PHI4[w]]);
    apply_group(sR, sI, g, gbA, rbA, lane);
#pragma unroll
    for (int w = 0; w < 4; ++w) {
        int q = 4 + w;
        float eta = (q == 4) ? 0.f : xb[ETA4[q]];   // MET has no eta
        g[w] = enc_gate(eta, xb[PT4[q]], xb[PHI4[q]]);
    }
    apply_group(sR, sI, g, gbB, rbB, lane);
    g[0] = ident_gate(); g[1] = ident_gate();
    g[2] = enc_gate(xb[ETA4[8]], xb[PT4[8]], xb[PHI4[8]]);
    g[3] = enc_gate(xb[ETA4[9]], xb[PT4[9]], xb[PHI4[9]]);
    apply_group(sR, sI, g, gbC, rbC, lane);
    cnot_chain10(sR, sI, lane);

    // ---- variational layers: per-wire RY*RX, grouped ----
    for (int d = 0; d < 2; ++d) {
        const float* wd = w4 + d * 20;
#pragma unroll
        for (int w = 0; w < 4; ++w) g[w] = layer_gate(wd[w], wd[10 + w]);
        apply_group(sR, sI, g, gbA, rbA, lane);
#pragma unroll
        for (int w = 0; w < 4; ++w) g[w] = layer_gate(wd[4 + w], wd[14 + w]);
        apply_group(sR, sI, g, gbB, rbB, lane);
        g[0] = ident_gate(); g[1] = ident_gate();
        g[2] = layer_gate(wd[8], wd[18]);
        g[3] = layer_gate(wd[9], wd[19]);
        apply_group(sR, sI, g, gbC, rbC, lane);
        cnot_chain10(sR, sI, lane);
    }

    measure10(sR, sI, lane, out + b * 32);
}

// ------------------------- blocks 1-3: per-thread --------------------------

__device__ __forceinline__ void apply1(C2& a0, C2& a1, const M2& M) {
    C2 n0 = { M.m[0][0].r * a0.r - M.m[0][0].i * a0.i + M.m[0][1].r * a1.r - M.m[0][1].i * a1.i,
              M.m[0][0].r * a0.i + M.m[0][0].i * a0.r + M.m[0][1].r * a1.i + M.m[0][1].i * a1.r };
    C2 n1 = { M.m[1][0].r * a0.r - M.m[1][0].i * a0.i + M.m[1][1].r * a1.r - M.m[1][1].i * a1.i,
              M.m[1][0].r * a0.i + M.m[1][0].i * a0.r + M.m[1][1].r * a1.i + M.m[1][1].i * a1.r };
    a0 = n0; a1 = n1;
}

__device__ __forceinline__ void apply4(float* sr, float* si, const M2& M, int bit) {
    const int mask = 1 << bit;
#pragma unroll
    for (int j = 0; j < 16; ++j) {
        if (j & mask) continue;
        int j1 = j | mask;
        float ar = sr[j], ai = si[j], br = sr[j1], bi = si[j1];
        sr[j]  = M.m[0][0].r * ar - M.m[0][0].i * ai + M.m[0][1].r * br - M.m[0][1].i * bi;
        si[j]  = M.m[0][0].r * ai + M.m[0][0].i * ar + M.m[0][1].r * bi + M.m[0][1].i * br;
        sr[j1] = M.m[1][0].r * ar - M.m[1][0].i * ai + M.m[1][1].r * br - M.m[1][1].i * bi;
        si[j1] = M.m[1][0].r * ai + M.m[1][0].i * ar + M.m[1][1].r * bi + M.m[1][1].i * br;
    }
}

__device__ __forceinline__ void cnots4(float* sr, float* si) {
#pragma unroll
    for (int c = 0; c < 3; ++c) {
        int cb = 3 - c, tb = 2 - c;
#pragma unroll
        for (int j = 0; j < 16; ++j) {
            if (((j >> cb) & 1) && !((j >> tb) & 1)) {
                int j2 = j ^ (1 << tb);
                float t1 = sr[j]; sr[j] = sr[j2]; sr[j2] = t1;
                float t2 = si[j]; si[j] = si[j2]; si[j2] = t2;
            }
        }
    }
}

__device__ __forceinline__ void sim1(const float* xb, const float* w1, float* outp) {
    C2 a0 = { 1.f, 0.f }, a1 = { 0.f, 0.f };
    M2 M = enc_gate(xb[21], xb[17], xb[25]);   // RX(eta), RY(pt), RZ(phi)
    apply1(a0, a1, M);
    for (int d = 0; d < 2; ++d) {
        M2 L = enc_gate(w1[3 * d], w1[3 * d + 1], w1[3 * d + 2]);  // RX,RY,RZ
        apply1(a0, a1, L);
    }
    float z  = a0.r * a0.r + a0.i * a0.i - a1.r * a1.r - a1.i * a1.i;
    float pr = a0.r * a1.r + a0.i * a1.i;
    float pi = a0.r * a1.i - a0.i * a1.r;
    outp[0] = z; outp[1] = 2.f * pr; outp[2] = 2.f * pi;
}

template <bool WITH_X0>
__device__ __forceinline__ void sim4(const float* xb, const float* wgt,
                                     const int* PT, const int* ETA, const int* PHI,
                                     float* outp) {
    float sr[16], si[16];
#pragma unroll
    for (int j = 0; j < 16; ++j) { sr[j] = 0.f; si[j] = 0.f; }
    sr[0] = 1.f;
#pragma unroll
    for (int k = 0; k < 4; ++k) {
        M2 M = enc_gate(xb[ETA[k]], xb[PT[k]], xb[PHI[k]]);
        apply4(sr, si, M, 3 - k);
    }
    cnots4(sr, si);
    for (int d = 0; d < 2; ++d) {
#pragma unroll
        for (int k = 0; k < 4; ++k) {
            M2 M = layer_gate(wgt[d * 8 + k], wgt[d * 8 + 4 + k]);
            apply4(sr, si, M, 3 - k);
        }
        cnots4(sr, si);
    }
#pragma unroll
    for (int k = 0; k < 4; ++k) {
        float z = 0.f;
#pragma unroll
        for (int j = 0; j < 16; ++j) {
            float p = sr[j] * sr[j] + si[j] * si[j];
            z += ((j >> (3 - k)) & 1) ? -p : p;
        }
        outp[k] = z;
    }
    if (WITH_X0) {
        float xr = 0.f;
#pragma unroll
        for (int j = 0; j < 8; ++j) xr += sr[j] * sr[j + 8] + si[j] * si[j + 8];
        outp[4] = 2.f * xr;
    }
}

__global__ void __launch_bounds__(256)
qenc_blocks123(const float* __restrict__ x, const float* __restrict__ w1,
               const float* __restrict__ w2, const float* __restrict__ w3,
               float* __restrict__ out) {
    int b = blockIdx.x * blockDim.x + threadIdx.x;
    const float* xb = x + b * 56;
    float* o = out + b * 32;
    sim1(xb, w1, o);
    const int PT2[4]  = { 5, 4, 35, 34 },  ETA2[4] = { 9, 8, 45, 44 },  PHI2[4] = { 13, 12, 55, 54 };
    const int PT3[4]  = { 28, 32, 15, 16 }, ETA3[4] = { 38, 42, 19, 20 }, PHI3[4] = { 48, 52, 23, 24 };
    sim4<false>(xb, w2, PT2, ETA2, PHI2, o + 3);
    sim4<true>(xb, w3, PT3, ETA3, PHI3, o + 7);
}

// ---------------------------------------------------------------------------

extern "C" void kernel_launch(void* const* d_in, const int* in_sizes, int n_in,
                              void* d_out, int out_size, void* d_ws, size_t ws_size,
                              hipStream_t stream) {
    const float* x  = (const float*)d_in[0];
    const float* w1 = (const float*)d_in[1];
    const float* w2 = (const float*)d_in[2];
    const float* w3 = (const float*)d_in[3];
    const float* w4 = (const float*)d_in[4];
    float* out = (float*)d_out;
    (void)in_sizes; (void)n_in; (void)out_size; (void)d_ws; (void)ws_size;

    // 2048 samples: cols 0..11 from blocks 1-3, cols 12..31 from block 4.
    qenc_blocks123<<<dim3(8), dim3(256), 0, stream>>>(x, w1, w2, w3, out);
    qenc_block4<<<dim3(512), dim3(128), 0, stream>>>(x, w4, out);
}